// AdultConnectome_10299331576311
// MI455X (gfx1250) — compile-verified
//
#include <hip/hip_runtime.h>
#include <hip/hip_bf16.h>
#include <stdint.h>

// y = A^5 x, COO SpMM, memory/atomic bound (AI ~0.1 FLOP/B).
// CDNA5 paths: global_load_async_to_lds_b32 (ASYNCcnt) edge staging,
// explicit global_atomic_add_f32 scope:SCOPE_DEV scatter, ds_load_b128
// edge-triple broadcast, b128 gathers of the L2-resident feature table.

#define N_FEAT  512
#define EPB     2048   // edges per block
#define CHUNK   256    // edges per async-staged chunk (== blockDim.x)
#define NCHUNK  (EPB / CHUNK)

// ---------- CDNA5 async global->LDS staging ----------
__device__ __forceinline__ void async_ld_b32(const void* lds_ptr, const void* gptr) {
  // GLOBAL_LOAD_ASYNC_TO_LDS_B32, GV mode. VDST = LDS byte addr, VADDR = 64b global.
  uint32_t loff = (uint32_t)(uintptr_t)lds_ptr;  // low 32 bits of generic ptr = LDS offset
  asm volatile("global_load_async_to_lds_b32 %0, %1, off"
               :: "v"(loff), "v"(gptr)
               : "memory");
}

__device__ __forceinline__ void wait_async0() {
  asm volatile("s_wait_asynccnt 0" ::: "memory");
}

// ---------- guaranteed-hardware fp32 scatter-add (device scope) ----------
__device__ __forceinline__ void atomic_fadd4(float* p, float a, float b, float c, float d) {
  // Non-returning GLOBAL_ATOMIC_ADD_F32 (STOREcnt); RMW resolves in L2 atomic units.
  asm volatile(
      "global_atomic_add_f32 %0, %1, off scope:SCOPE_DEV\n\t"
      "global_atomic_add_f32 %0, %2, off offset:4 scope:SCOPE_DEV\n\t"
      "global_atomic_add_f32 %0, %3, off offset:8 scope:SCOPE_DEV\n\t"
      "global_atomic_add_f32 %0, %4, off offset:12 scope:SCOPE_DEV"
      :: "v"(p), "v"(a), "v"(b), "v"(c), "v"(d)
      : "memory");
}

// ---------- one SpMM hop: hout[row[e],:] += w[e] * hin[col[e],:] ----------
__global__ __launch_bounds__(256)
void spmm_hop(const int*   __restrict__ row,
              const int*   __restrict__ col,
              const float* __restrict__ wgt,
              const float* __restrict__ hin,
              float*       __restrict__ hout,
              int nedges) {
  // One 16B slot per edge: [row, col, w_bits, pad] -> single ds_load_b128 broadcast.
  __shared__ int s_edge[2][CHUNK][4];

  const int  tid  = threadIdx.x;
  const int  half = tid >> 7;          // block split into two 128-lane halves
  const int  f0   = (tid & 127) * 4;   // 128 lanes x float4 = 512 features
  const long blockBase = (long)blockIdx.x * EPB;

  // Prologue: stage chunk 0 into LDS buffer 0.
  {
    long g = blockBase + tid;
    if (g < nedges) {
      async_ld_b32(&s_edge[0][tid][0], row + g);
      async_ld_b32(&s_edge[0][tid][1], col + g);
      async_ld_b32(&s_edge[0][tid][2], wgt + g);
    }
  }

  for (int c = 0; c < NCHUNK; ++c) {
    const int buf = c & 1;

    // Wait for this wave's async loads into buf, then barrier: makes the DMA'd
    // LDS visible block-wide and guarantees nobody still reads the other buffer.
    wait_async0();
    __syncthreads();

    // Stage next chunk into the other buffer (overlaps with compute below).
    if (c + 1 < NCHUNK) {
      long g = blockBase + (long)(c + 1) * CHUNK + tid;
      if (g < nedges) {
        const int nb = buf ^ 1;
        async_ld_b32(&s_edge[nb][tid][0], row + g);
        async_ld_b32(&s_edge[nb][tid][1], col + g);
        async_ld_b32(&s_edge[nb][tid][2], wgt + g);
      }
    }

    long cbase = blockBase + (long)c * CHUNK;
    int  cnt   = (int)(nedges - cbase);
    if (cnt > CHUNK) cnt = CHUNK;
    if (cnt < 0)     cnt = 0;

    // Each 128-lane half processes one edge per iteration:
    // 1x ds_load_b128 (edge triple, broadcast) + 1x b128 gather + 4x f32 atomics.
    for (int j = 0; j < cnt; j += 2) {
      int e = j + half;
      if (e < cnt) {
        const int4 t = *reinterpret_cast<const int4*>(&s_edge[buf][e][0]);
        const int   r  = t.x;
        const int   cc = t.y;
        const float wv = __int_as_float(t.z);
        const float4 hv =
            *reinterpret_cast<const float4*>(hin + (size_t)cc * N_FEAT + f0);
        float* op = hout + (size_t)r * N_FEAT + f0;
        atomic_fadd4(op, wv * hv.x, wv * hv.y, wv * hv.z, wv * hv.w);
      }
    }
    // No trailing barrier: next iteration's wait+barrier orders DMA writes vs
    // readers, and compute(buf) vs the following DMA into buf.
  }
}

// ---------- zero-fill (accumulator init each hop) ----------
__global__ void zero_f4(float4* __restrict__ p, long n4) {
  long i      = (long)blockIdx.x * blockDim.x + threadIdx.x;
  long stride = (long)gridDim.x * blockDim.x;
  float4 z = {0.f, 0.f, 0.f, 0.f};
  for (; i < n4; i += stride) p[i] = z;
}

extern "C" void kernel_launch(void* const* d_in, const int* in_sizes, int n_in,
                              void* d_out, int out_size, void* d_ws, size_t ws_size,
                              hipStream_t stream) {
  (void)n_in; (void)out_size; (void)ws_size;

  const float* x   = (const float*)d_in[0];  // (N_NODES, 512) f32
  const float* wgt = (const float*)d_in[1];  // (E,) f32
  const int*   row = (const int*)  d_in[2];  // (E,) int
  const int*   col = (const int*)  d_in[3];  // (E,) int
  const int nedges = in_sizes[1];
  const int nnodes = in_sizes[0] / N_FEAT;
  const int L = 5;  // layer_number fixed by setup_inputs()

  float* out = (float*)d_out;
  float* ws  = (float*)d_ws;   // ping-pong buffer: nnodes*512 floats (204.8 MB)

  const long nf = (long)nnodes * N_FEAT;
  const int  zthreads = 256;
  const int  zblocks  = (int)((nf / 4 + zthreads - 1) / zthreads);
  const int  sblocks  = (nedges + EPB - 1) / EPB;

  // Ping-pong between {ws, out} so that hop L lands in d_out:
  // dst(k) = out if (L-k) even, else ws.
  const float* src = x;
  for (int k = 1; k <= L; ++k) {
    float* dst = (((L - k) & 1) == 0) ? out : ws;
    zero_f4<<<zblocks, zthreads, 0, stream>>>((float4*)dst, nf / 4);
    spmm_hop<<<sblocks, 256, 0, stream>>>(row, col, wgt, src, dst, nedges);
    src = dst;
  }
}